// LightGCN_V8_27307402068157
// MI455X (gfx1250) — compile-verified
//
#include <hip/hip_runtime.h>

typedef __attribute__((ext_vector_type(2))) float v2f;
typedef __attribute__((ext_vector_type(8))) float v8f;

#define EMB 64
#define SCAN_T 1024

// ---- integer degree count over targets: cnt[col[e]] += 1 ----------------
__global__ void lgcn_count_kernel(const int* __restrict__ col, int* __restrict__ cnt,
                                  int n_edges) {
    int e = blockIdx.x * blockDim.x + threadIdx.x;
    if (e < n_edges) atomicAdd(&cnt[col[e]], 1);
}

// ---- dis[i] = cnt>0 ? rsqrt(cnt) : 0 ------------------------------------
__global__ void lgcn_dis_kernel(const int* __restrict__ cnt, float* __restrict__ dis,
                                int n_nodes) {
    int i = blockIdx.x * blockDim.x + threadIdx.x;
    if (i < n_nodes) {
        int c = cnt[i];
        dis[i] = (c > 0) ? rsqrtf((float)c) : 0.0f;
    }
}

// ---- exclusive scan of cnt -> offs (single block, chunked LDS scan) -----
__global__ void lgcn_scan_kernel(const int* __restrict__ cnt, int* __restrict__ offs,
                                 int n_nodes) {
    __shared__ int sdata[SCAN_T];
    __shared__ int carry_s;
    const int tid = threadIdx.x;
    if (tid == 0) carry_s = 0;
    __syncthreads();
    for (int base = 0; base < n_nodes; base += SCAN_T) {
        int i = base + tid;
        int v = (i < n_nodes) ? cnt[i] : 0;
        sdata[tid] = v;
        __syncthreads();
#pragma unroll
        for (int ofs = 1; ofs < SCAN_T; ofs <<= 1) {   // Hillis-Steele inclusive
            int t = (tid >= ofs) ? sdata[tid - ofs] : 0;
            __syncthreads();
            sdata[tid] += t;
            __syncthreads();
        }
        int carry = carry_s;
        if (i < n_nodes) offs[i] = carry + sdata[tid] - v;   // exclusive
        __syncthreads();
        if (tid == SCAN_T - 1) carry_s = carry + sdata[tid];
        __syncthreads();
    }
    if (tid == 0) offs[n_nodes] = carry_s;   // == n_edges
}

// ---- CSR fill: pack[pos] = {row[e], norm[e]} grouped by col --------------
__global__ void lgcn_fill_kernel(const int* __restrict__ row, const int* __restrict__ col,
                                 const float* __restrict__ dis, const int* __restrict__ offs,
                                 int* __restrict__ cursor, int2* __restrict__ pack,
                                 int n_edges) {
    int e = blockIdx.x * blockDim.x + threadIdx.x;
    if (e >= n_edges) return;
    int r = row[e];
    int c = col[e];
    float nm = dis[r] * dis[c];
    int pos = offs[c] + atomicAdd(&cursor[c], 1);
    pack[pos] = make_int2(r, __float_as_int(nm));
}

// ---- acc = x, total = x (float4 vectorized) ------------------------------
__global__ void lgcn_init_kernel(const float* __restrict__ uw, const float* __restrict__ iw,
                                 float* __restrict__ acc, float* __restrict__ total,
                                 int n_users, long long n4) {
    long long i = (long long)blockIdx.x * blockDim.x + threadIdx.x;
    if (i >= n4) return;
    long long node = i >> 4;  // 16 float4s per node
    float4 v;
    if (node < n_users)
        v = ((const float4*)uw)[i];
    else
        v = ((const float4*)iw)[i - (long long)n_users * 16];
    ((float4*)acc)[i]   = v;
    ((float4*)total)[i] = v;
}

// ---- one wave per target node: atomic-free gather layer ------------------
//   acc_out[c,:] = sum_{e: col==c} norm_e * acc_in[row_e,:]
//   total[c,:]  += acc_out[c,:]            (fused; wave owns the node)
// Edge records are wave-uniform (scalar path); gathers are 256B coalesced.
__global__ void lgcn_layer_kernel(const int2* __restrict__ pack, const int* __restrict__ offs,
                                  const float* __restrict__ acc_in,
                                  float* __restrict__ acc_out, float* __restrict__ total,
                                  int n_nodes) {
    const int lane = threadIdx.x & 31;
    const int node = __builtin_amdgcn_readfirstlane(
        (int)blockIdx.x * (int)(blockDim.x >> 5) + (int)(threadIdx.x >> 5));
    if (node >= n_nodes) return;              // uniform per wave
    const int beg = offs[node];
    const int end = offs[node + 1];
    float sx = 0.0f, sy = 0.0f;
    for (int k = beg; k < end; ++k) {
        const int2 p = pack[k];               // uniform -> scalar load
        const float nm = __int_as_float(p.y);
        const float2 v = *(const float2*)(acc_in + (long long)p.x * EMB + lane * 2);
        sx = fmaf(nm, v.x, sx);
        sy = fmaf(nm, v.y, sy);
    }
    const long long o = (long long)node * EMB + lane * 2;
    *(float2*)(acc_out + o) = make_float2(sx, sy);
    float2 t = *(float2*)(total + o);
    *(float2*)(total + o) = make_float2(t.x + sx, t.y + sy);
}

// ---- finalize: out *= 0.25 via V_WMMA_F32_16X16X4_F32, one wave/tile -----
// D(16x16) = sum_{j=0..3} A_j(16x4) x B_j(4x16), A_j = tile columns 4j..4j+3,
// B_j[k][n] = 0.25 * delta(n, 4j+k). All f32; power-of-two scale is exact.
__global__ void lgcn_finalize_wmma(float* __restrict__ out, int n_nodes) {
    const int lane = threadIdx.x & 31;
    const int tile = blockIdx.x * (blockDim.x >> 5) + (threadIdx.x >> 5);
    const int n_tiles = (n_nodes >> 4) * (EMB / 16);
    if (tile >= n_tiles) return;   // uniform per wave -> EXEC all-ones for WMMA
    const int rt = tile >> 2;
    const int ct = tile & 3;
    const int m  = lane & 15;
    const int h  = lane >> 4;

    const long long rowBase = (long long)(rt * 16 + m) * EMB + ct * 16;

    v8f d = {};
#pragma unroll
    for (int j = 0; j < 4; ++j) {
        v2f a;
        a.x = out[rowBase + j * 4 + 2 * h + 0];
        a.y = out[rowBase + j * 4 + 2 * h + 1];
        v2f b;
        b.x = (m == (j * 4 + 2 * h + 0)) ? 0.25f : 0.0f;
        b.y = (m == (j * 4 + 2 * h + 1)) ? 0.25f : 0.0f;
        d = __builtin_amdgcn_wmma_f32_16x16x4_f32(
                false, a, false, b, (short)0, d, false, false);
    }
#pragma unroll
    for (int r = 0; r < 8; ++r) {
        out[(long long)(rt * 16 + r + 8 * h) * EMB + ct * 16 + m] = d[r];
    }
}

extern "C" void kernel_launch(void* const* d_in, const int* in_sizes, int n_in,
                              void* d_out, int out_size, void* d_ws, size_t ws_size,
                              hipStream_t stream) {
    const int*   edge = (const int*)d_in[0];
    const float* uw   = (const float*)d_in[1];
    const float* iw   = (const float*)d_in[2];

    const int n_edges = in_sizes[0] / 2;
    const int n_users = in_sizes[1] / EMB;
    const int n_items = in_sizes[2] / EMB;
    const int n_nodes = n_users + n_items;

    const int* row = edge;            // edge_index[0]
    const int* col = edge + n_edges;  // edge_index[1]

    // workspace: cnt | cursor | dis | offs | pack | accA | accB  (~111 MB)
    char* ws = (char*)d_ws;
    auto al = [](size_t x) { return (x + 255) & ~(size_t)255; };
    size_t off = 0;
    int*   cnt    = (int*)(ws + off);  off = al(off + (size_t)n_nodes * 4);
    int*   cursor = (int*)(ws + off);  off = al(off + (size_t)n_nodes * 4);
    float* dis    = (float*)(ws + off); off = al(off + (size_t)n_nodes * 4);
    int*   offs   = (int*)(ws + off);  off = al(off + ((size_t)n_nodes + 1) * 4);
    int2*  pack   = (int2*)(ws + off); off = al(off + (size_t)n_edges * 8);
    float* accA   = (float*)(ws + off); off = al(off + (size_t)n_nodes * EMB * 4);
    float* accB   = (float*)(ws + off);
    float* total  = (float*)d_out;

    const long long n4 = (long long)n_nodes * (EMB / 4);
    const int TB = 256;

    // 1) CSR build + normalization (integer atomics only)
    hipMemsetAsync(cnt, 0, (size_t)n_nodes * 4, stream);
    hipMemsetAsync(cursor, 0, (size_t)n_nodes * 4, stream);
    lgcn_count_kernel<<<(n_edges + TB - 1) / TB, TB, 0, stream>>>(col, cnt, n_edges);
    lgcn_dis_kernel  <<<(n_nodes + TB - 1) / TB, TB, 0, stream>>>(cnt, dis, n_nodes);
    lgcn_scan_kernel <<<1, SCAN_T, 0, stream>>>(cnt, offs, n_nodes);
    lgcn_fill_kernel <<<(n_edges + TB - 1) / TB, TB, 0, stream>>>(
        row, col, dis, offs, cursor, pack, n_edges);

    // 2) acc = x, total = x
    lgcn_init_kernel<<<(unsigned)((n4 + TB - 1) / TB), TB, 0, stream>>>(
        uw, iw, accA, total, n_users, n4);

    // 3) three atomic-free gather layers (ping-pong), total += fused
    float* cur = accA;
    float* nxt = accB;
    const unsigned layer_blocks = (unsigned)(((long long)n_nodes * 32 + TB - 1) / TB);
    for (int l = 0; l < 3; ++l) {
        lgcn_layer_kernel<<<layer_blocks, TB, 0, stream>>>(pack, offs, cur, nxt, total, n_nodes);
        float* t = cur; cur = nxt; nxt = t;
    }

    // 4) total *= 0.25 via WMMA (exact power-of-two scale)
    const int n_tiles = (n_nodes / 16) * (EMB / 16);
    lgcn_finalize_wmma<<<(n_tiles + 7) / 8, TB, 0, stream>>>(total, n_nodes);
}